// MultiPosCrossEntropyLoss_66048007078606
// MI455X (gfx1250) — compile-verified
//
#include <hip/hip_runtime.h>
#include <hip/hip_bf16.h>

// Problem constants from the reference.
#define N_ROWS 8192
#define N_COLS 4096
#define TILE_ROWS 16
#define WAVES_PER_BLOCK 8
#define COLS_PER_WAVE (N_COLS / WAVES_PER_BLOCK)   // 512
#define ITERS_PER_WAVE (COLS_PER_WAVE / 8)         // 64 (8 cols/row per iter)

typedef float v2f  __attribute__((ext_vector_type(2)));
typedef float v8f  __attribute__((ext_vector_type(8)));
typedef float f32x4 __attribute__((ext_vector_type(4)));
typedef int   i32x4 __attribute__((ext_vector_type(4)));

// D = A(16x4,f32) x B(4x16,f32) + C(16x16,f32).  With B = ones,
// D[m,n] += sum_k A[m,k]  -> exact f32 per-row K-reduction on the matrix pipe.
__device__ __forceinline__ v8f wmma_rowsum_acc(v2f a, v8f c) {
    const v2f ones = {1.0f, 1.0f};
    return __builtin_amdgcn_wmma_f32_16x16x4_f32(
        /*neg_a=*/false, a, /*neg_b=*/false, ones,
        /*c_mod=*/(short)0, c, /*reuse_a=*/false, /*reuse_b=*/false);
}

// Kernel 1: one block per 16-row tile.  8 waves split the 4096 columns.
// Per element: e = exp(label ? -x : x); route e into the neg- or pos-
// accumulator; each WMMA contracts K=4 masked-exp values into C (f32, RNE).
// Each lane streams float4/int4 (B128, NT hint) and feeds two WMMAs per
// accumulator per load.
__global__ void __launch_bounds__(256)
mpce_row_loss_kernel(const float* __restrict__ cls, const int* __restrict__ lab,
                     float* __restrict__ row_loss) {
    __shared__ float pN[WAVES_PER_BLOCK][TILE_ROWS];
    __shared__ float pP[WAVES_PER_BLOCK][TILE_ROWS];

    const int tid  = threadIdx.x;
    const int w    = tid >> 5;        // wave id 0..7
    const int lane = tid & 31;
    const int r    = lane & 15;       // tile row this lane feeds
    const int hi   = lane >> 4;       // 0 -> K={0,1}, 1 -> K={2,3}
    const int row0 = blockIdx.x * TILE_ROWS;

    const float* xrow = cls + (size_t)(row0 + r) * N_COLS;
    const int*   lrow = lab + (size_t)(row0 + r) * N_COLS;
    const int colBase = w * COLS_PER_WAVE + 4 * hi;

    v8f cn = {};   // sum over neg of exp(x)
    v8f cp = {};   // sum over pos of exp(-x)

#pragma unroll 4
    for (int it = 0; it < ITERS_PER_WAVE; ++it) {
        const int col = colBase + it * 8;
        const f32x4 x  = __builtin_nontemporal_load((const f32x4*)(xrow + col));
        const i32x4 lb = __builtin_nontemporal_load((const i32x4*)(lrow + col));

        // one exp per element; each element is exclusively pos or neg
        const bool p0 = (lb.x != 0), p1 = (lb.y != 0);
        const bool p2 = (lb.z != 0), p3 = (lb.w != 0);
        const float e0 = __expf(p0 ? -x.x : x.x);
        const float e1 = __expf(p1 ? -x.y : x.y);
        const float e2 = __expf(p2 ? -x.z : x.z);
        const float e3 = __expf(p3 ? -x.w : x.w);

        // first half of the float4 -> one K=4 tile, second half -> another
        v2f anA = { p0 ? 0.0f : e0, p1 ? 0.0f : e1 };
        v2f apA = { p0 ? e0 : 0.0f, p1 ? e1 : 0.0f };
        v2f anB = { p2 ? 0.0f : e2, p3 ? 0.0f : e3 };
        v2f apB = { p2 ? e2 : 0.0f, p3 ? e3 : 0.0f };

        cn = wmma_rowsum_acc(anA, cn);
        cp = wmma_rowsum_acc(apA, cp);
        cn = wmma_rowsum_acc(anB, cn);
        cp = wmma_rowsum_acc(apB, cp);
    }

    // D[m,n] is the row-m sum broadcast over n.  Lane 0 holds rows 0..7 in
    // c[0..7]; lane 16 holds rows 8..15 in c[0..7].
    if (lane == 0) {
#pragma unroll
        for (int k = 0; k < 8; ++k) { pN[w][k] = cn[k]; pP[w][k] = cp[k]; }
    } else if (lane == 16) {
#pragma unroll
        for (int k = 0; k < 8; ++k) { pN[w][8 + k] = cn[k]; pP[w][8 + k] = cp[k]; }
    }
    __syncthreads();

    if (tid < TILE_ROWS) {
        float sn = 0.0f, sp = 0.0f;
#pragma unroll
        for (int ww = 0; ww < WAVES_PER_BLOCK; ++ww) { sn += pN[ww][tid]; sp += pP[ww][tid]; }
        // loss = logaddexp(0, log(sn) + log(sp)) = log(1 + sn*sp)
        row_loss[row0 + tid] = __logf(1.0f + sn * sp);
    }
}

// Kernel 2: one wave reduces the 8192 row losses to the mean.
// A = ones, B = 64 contiguous values/iter -> D[m,n] += column sums (f32).
__global__ void __launch_bounds__(32)
mpce_mean_kernel(const float* __restrict__ row_loss, float* __restrict__ out) {
    const int lane = threadIdx.x & 31;
    const v2f ones = {1.0f, 1.0f};
    v8f c = {};

    for (int i = 0; i < N_ROWS; i += 64) {
        v2f b = { row_loss[i + lane], row_loss[i + 32 + lane] };
        c = __builtin_amdgcn_wmma_f32_16x16x4_f32(
            false, ones, false, b, (short)0, c, false, false);
    }

    // c[0] lane n (0..15) = colsum_n; lanes 16..31 duplicate.  Sum the 16-lane half.
    float v = c[0];
    v += __shfl_xor(v, 1, 32);
    v += __shfl_xor(v, 2, 32);
    v += __shfl_xor(v, 4, 32);
    v += __shfl_xor(v, 8, 32);

    if (threadIdx.x == 0) {
        out[0] = v * (1.0f / (float)N_ROWS);   // LOSS_WEIGHT == 1.0
    }
}

extern "C" void kernel_launch(void* const* d_in, const int* in_sizes, int n_in,
                              void* d_out, int out_size, void* d_ws, size_t ws_size,
                              hipStream_t stream) {
    const float* cls = (const float*)d_in[0];
    const int*   lab = (const int*)d_in[1];
    float* out = (float*)d_out;
    float* ws  = (float*)d_ws;   // N_ROWS per-row losses (32 KB)

    mpce_row_loss_kernel<<<N_ROWS / TILE_ROWS, 256, 0, stream>>>(cls, lab, ws);
    mpce_mean_kernel<<<1, 32, 0, stream>>>(ws, out);
}